// NeuralNet_21337397526971
// MI455X (gfx1250) — compile-verified
//
#include <hip/hip_runtime.h>

typedef __attribute__((ext_vector_type(2))) float v2f;
typedef __attribute__((ext_vector_type(8))) float v8f;

#define BATCH 16384
#define EPS 1e-5f

// ---------------------------------------------------------------------------
// Head GEMM weight construction.
// Effective weight Wz[k][n], k = i*52 + j (i in 0..49, j in 0..51), n in 0..15:
//   n >= 10           -> 0          (N padding)
//   j >= 49           -> 0          (K row padding, multiple-of-4 alignment)
//   i <  49           -> (i<=j) ? w_fc2[n,i*49+j] : 0   (triu mask baked in)
//   i == 49           -> w_fc1[n,j]                      (linear term, A row = 1)
// Stored pre-swizzled for the wave: at K-step s, lane l holds float2
// { Wz[4s+2*(l/16)][l%16], Wz[4s+2*(l/16)+1][l%16] }  -> B VGPR0/VGPR1.
// ---------------------------------------------------------------------------
__device__ inline float wz_val(const float* w_fc1, const float* w_fc2, int k, int n) {
  if (n >= 10) return 0.f;
  int i = k / 52, j = k - i * 52;
  if (j >= 49) return 0.f;
  if (i < 49) return (i <= j) ? w_fc2[n * 2401 + i * 49 + j] : 0.f;
  return w_fc1[n * 49 + j];
}

__global__ void prep_kernel(const float* __restrict__ w_fc1,
                            const float* __restrict__ w_fc2,
                            float2* __restrict__ wswz,
                            float* __restrict__ stats) {
  int t = blockIdx.x * blockDim.x + threadIdx.x;
  if (t < 4) stats[t] = 0.f;                 // zero BN accumulators each call
  if (t < 650 * 32) {
    int s = t >> 5, lane = t & 31;
    int n = lane & 15, half = lane >> 4;
    int ka = 4 * s + 2 * half;
    float2 v;
    v.x = wz_val(w_fc1, w_fc2, ka, n);
    v.y = wz_val(w_fc1, w_fc2, ka + 1, n);
    wswz[t] = v;
  }
}

// prods order from reference: [b00^2,b01^2,b10^2,b11^2,b00b01,b00b10,b00b11,b01b10,b01b11,b10b11]
__device__ inline float dot10(float t00, float t01, float t10, float t11,
                              const float* __restrict__ W) {
  return W[0]*t00*t00 + W[1]*t01*t01 + W[2]*t10*t10 + W[3]*t11*t11
       + W[4]*t00*t01 + W[5]*t00*t10 + W[6]*t00*t11
       + W[7]*t01*t10 + W[8]*t01*t11 + W[9]*t10*t11;
}

// ---------------------------------------------------------------------------
// Layer 1: one wave per image (8 images / 256-thread block).
// l1[r,c] = pool22(x*w1) + sum_{4 branches} dot10(masked blocks, w2_k[r,10c..]) + w3
// Accumulates BN sum / sumsq into stats[0..1].
// ---------------------------------------------------------------------------
__global__ __launch_bounds__(256) void layer1_kernel(
    const float* __restrict__ x,
    const float* __restrict__ w1,   const float* __restrict__ w2lr,
    const float* __restrict__ w2tb, const float* __restrict__ w2all,
    const float* __restrict__ w2no, const float* __restrict__ w3,
    float* __restrict__ l1_out, float* __restrict__ stats) {
  __shared__ float s_w1[784];
  __shared__ float s_w2[4][1960];
  __shared__ float s_w3[196];
  __shared__ float s_img[8][784];
  __shared__ float s_red[256];
  int tid = threadIdx.x;
  for (int p = tid; p < 784; p += 256) s_w1[p] = w1[p];
  for (int p = tid; p < 1960; p += 256) {
    s_w2[0][p] = w2lr[p]; s_w2[1][p] = w2tb[p];
    s_w2[2][p] = w2all[p]; s_w2[3][p] = w2no[p];
  }
  for (int p = tid; p < 196; p += 256) s_w3[p] = w3[p];
  int wv = tid >> 5, lane = tid & 31;
  long b = (long)blockIdx.x * 8 + wv;
  const float* xb = x + b * 784;
  for (int p = lane; p < 784; p += 32) s_img[wv][p] = xb[p];
  __syncthreads();

  float sum = 0.f, sq = 0.f;
  for (int pix = lane; pix < 196; pix += 32) {
    int r = pix / 14, c = pix - r * 14;
    int i00 = (2 * r) * 28 + 2 * c;
    float b00 = s_img[wv][i00],      b01 = s_img[wv][i00 + 1];
    float b10 = s_img[wv][i00 + 28], b11 = s_img[wv][i00 + 29];
    float acc = b00 * s_w1[i00] + b01 * s_w1[i00 + 1]
              + b10 * s_w1[i00 + 28] + b11 * s_w1[i00 + 29];
    acc += s_w3[pix];
    int wofs = r * 140 + 10 * c;
    { // left_right: cols 0 and 27 zeroed
      float t00 = (c == 0) ? 0.f : b00, t10 = (c == 0) ? 0.f : b10;
      float t01 = (c == 13) ? 0.f : b01, t11 = (c == 13) ? 0.f : b11;
      acc += dot10(t00, t01, t10, t11, &s_w2[0][wofs]);
    }
    { // top_bottom: rows 0 and 27 zeroed
      float t00 = (r == 0) ? 0.f : b00, t01 = (r == 0) ? 0.f : b01;
      float t10 = (r == 13) ? 0.f : b10, t11 = (r == 13) ? 0.f : b11;
      acc += dot10(t00, t01, t10, t11, &s_w2[1][wofs]);
    }
    { // all
      float t00 = (c == 0 || r == 0) ? 0.f : b00;
      float t01 = (c == 13 || r == 0) ? 0.f : b01;
      float t10 = (c == 0 || r == 13) ? 0.f : b10;
      float t11 = (c == 13 || r == 13) ? 0.f : b11;
      acc += dot10(t00, t01, t10, t11, &s_w2[2][wofs]);
    }
    acc += dot10(b00, b01, b10, b11, &s_w2[3][wofs]); // none
    l1_out[b * 196 + pix] = acc;
    sum += acc; sq += acc * acc;
  }
  s_red[tid] = sum; __syncthreads();
  for (int o = 128; o > 0; o >>= 1) { if (tid < o) s_red[tid] += s_red[tid + o]; __syncthreads(); }
  if (tid == 0) atomicAdd(&stats[0], s_red[0]);
  __syncthreads();
  s_red[tid] = sq; __syncthreads();
  for (int o = 128; o > 0; o >>= 1) { if (tid < o) s_red[tid] += s_red[tid + o]; __syncthreads(); }
  if (tid == 0) atomicAdd(&stats[1], s_red[0]);
}

// BN finalize: h = l * s + t with s = gamma*rsqrt(var+eps), t = beta - mean*s.
__global__ void stats_kernel(float* __restrict__ stats,
                             const float* __restrict__ gamma,
                             const float* __restrict__ beta,
                             float n, int sIdx, int oIdx) {
  float mean = stats[sIdx] / n;
  float var = stats[sIdx + 1] / n - mean * mean;
  float s = gamma[0] * rsqrtf(var + EPS);
  stats[oIdx] = s;
  stats[oIdx + 1] = beta[0] - mean * s;
}

// ---------------------------------------------------------------------------
// Layer 2: h = BN(l1); l2 = pool22(h*w1) + dot10(h blocks, w2) + w4 (7x7).
// ---------------------------------------------------------------------------
__global__ __launch_bounds__(256) void layer2_kernel(
    const float* __restrict__ l1_in,
    const float* __restrict__ w1, const float* __restrict__ w2,
    const float* __restrict__ w4,
    float* __restrict__ l2_out, float* __restrict__ stats) {
  __shared__ float s_w1[196];
  __shared__ float s_w2[490];
  __shared__ float s_w4[49];
  __shared__ float s_h[8][196];
  __shared__ float s_red[256];
  int tid = threadIdx.x;
  for (int p = tid; p < 196; p += 256) s_w1[p] = w1[p];
  for (int p = tid; p < 490; p += 256) s_w2[p] = w2[p];
  for (int p = tid; p < 49; p += 256) s_w4[p] = w4[p];
  float sbn = stats[4], tbn = stats[5];
  int wv = tid >> 5, lane = tid & 31;
  long b = (long)blockIdx.x * 8 + wv;
  const float* lb = l1_in + b * 196;
  for (int p = lane; p < 196; p += 32) s_h[wv][p] = lb[p] * sbn + tbn;
  __syncthreads();

  float sum = 0.f, sq = 0.f;
  for (int pix = lane; pix < 49; pix += 32) {
    int r = pix / 7, c = pix - r * 7;
    int i00 = (2 * r) * 14 + 2 * c;
    float b00 = s_h[wv][i00],      b01 = s_h[wv][i00 + 1];
    float b10 = s_h[wv][i00 + 14], b11 = s_h[wv][i00 + 15];
    float acc = b00 * s_w1[i00] + b01 * s_w1[i00 + 1]
              + b10 * s_w1[i00 + 14] + b11 * s_w1[i00 + 15];
    acc += s_w4[pix];
    acc += dot10(b00, b01, b10, b11, &s_w2[r * 70 + 10 * c]);
    l2_out[b * 49 + pix] = acc;
    sum += acc; sq += acc * acc;
  }
  s_red[tid] = sum; __syncthreads();
  for (int o = 128; o > 0; o >>= 1) { if (tid < o) s_red[tid] += s_red[tid + o]; __syncthreads(); }
  if (tid == 0) atomicAdd(&stats[2], s_red[0]);
  __syncthreads();
  s_red[tid] = sq; __syncthreads();
  for (int o = 128; o > 0; o >>= 1) { if (tid < o) s_red[tid] += s_red[tid + o]; __syncthreads(); }
  if (tid == 0) atomicAdd(&stats[3], s_red[0]);
}

// ---------------------------------------------------------------------------
// Head GEMM via V_WMMA_F32_16X16X4_F32: out = [outer(v,v)|v] @ Wz^T + bias.
// One wave = 16 batches (M=16), N=16 (10 used), K = 50 rows * 52 = 2600.
// A fragment (lane, step): row m=lane%16, K = 4*step + 2*(lane/16) + {0,1}.
// a(k) = v_i * v_j with k = i*52+j (v padded to 52 with zeros; i==49 -> v_i=1).
// ---------------------------------------------------------------------------
__global__ __launch_bounds__(256) void gemm_kernel(
    const float* __restrict__ l2_in, const float* __restrict__ stats,
    const float2* __restrict__ wswz,
    const float* __restrict__ b_fc1, const float* __restrict__ b_fc2,
    float* __restrict__ out) {
  __shared__ __align__(16) float s_v[8][16 * 52];
  int tid = threadIdx.x;
  int wv = tid >> 5, lane = tid & 31;
  int n = lane & 15, half = lane >> 4;
  long b0 = ((long)blockIdx.x * 8 + wv) * 16;
  float sbn = stats[6], tbn = stats[7];

  // Stage 16 BN'd vectors, zero-padded 49 -> 52 (832 = 26*32: no divergence).
  for (int idx = lane; idx < 16 * 52; idx += 32) {
    int m = idx / 52, j = idx - m * 52;
    float val = 0.f;
    if (j < 49) val = l2_in[(b0 + m) * 49 + j] * sbn + tbn;
    s_v[wv][idx] = val;
  }
  __syncthreads();

  const float* vp = s_v[wv] + n * 52;    // this lane's batch row
  v8f acc = {};
  const float2* wp = wswz + lane;
  for (int i = 0; i < 50; ++i) {
    float vi = (i < 49) ? vp[i] : 1.0f;
#pragma unroll
    for (int j0 = 0; j0 < 52; j0 += 4) {
      int j = j0 + 2 * half;
      float vj0 = vp[j], vj1 = vp[j + 1];
      v2f a; a.x = vi * vj0; a.y = vi * vj1;
      float2 w = *wp; wp += 32;
      v2f bm; bm.x = w.x; bm.y = w.y;
      acc = __builtin_amdgcn_wmma_f32_16x16x4_f32(
          false, a, false, bm, (short)0, acc, false, false);
    }
  }

  // C layout: VGPR r -> M=r (lanes 0-15) / M=8+r (lanes 16-31), N = n.
  if (n < 10) {
    float bias = b_fc1[n] + b_fc2[n];
#pragma unroll
    for (int r = 0; r < 8; ++r) {
      long M = b0 + r + 8 * half;
      out[M * 10 + n] = acc[r] + bias;
    }
  }
}

extern "C" void kernel_launch(void* const* d_in, const int* in_sizes, int n_in,
                              void* d_out, int out_size, void* d_ws, size_t ws_size,
                              hipStream_t stream) {
  (void)in_sizes; (void)n_in; (void)out_size; (void)ws_size;
  const float* x       = (const float*)d_in[0];
  const float* l1_w1   = (const float*)d_in[1];
  const float* l1_w2_1 = (const float*)d_in[2];
  const float* l1_w2_2 = (const float*)d_in[3];
  const float* l1_w2_3 = (const float*)d_in[4];
  const float* l1_w2_4 = (const float*)d_in[5];
  const float* l1_w3   = (const float*)d_in[6];
  const float* l2_w1   = (const float*)d_in[7];
  const float* l2_w2   = (const float*)d_in[8];
  const float* l2_w4   = (const float*)d_in[9];
  const float* w_fc1   = (const float*)d_in[10];
  const float* b_fc1   = (const float*)d_in[11];
  const float* w_fc2   = (const float*)d_in[12];
  const float* b_fc2   = (const float*)d_in[13];
  const float* gamma   = (const float*)d_in[14];
  const float* beta    = (const float*)d_in[15];
  float* out = (float*)d_out;

  float* l1_buf = (float*)d_ws;                       // 16384*196 f32
  float* l2_buf = l1_buf + (size_t)BATCH * 196;       // 16384*49 f32
  float* wswz   = l2_buf + (size_t)BATCH * 49;        // 650*32 float2
  float* stats  = wswz + (size_t)650 * 32 * 2;        // 8 f32

  prep_kernel<<<(650 * 32 + 255) / 256, 256, 0, stream>>>(w_fc1, w_fc2, (float2*)wswz, stats);
  layer1_kernel<<<BATCH / 8, 256, 0, stream>>>(x, l1_w1, l1_w2_1, l1_w2_2, l1_w2_3,
                                               l1_w2_4, l1_w3, l1_buf, stats);
  stats_kernel<<<1, 1, 0, stream>>>(stats, gamma, beta, (float)(BATCH * 196), 0, 4);
  layer2_kernel<<<BATCH / 8, 256, 0, stream>>>(l1_buf, l2_w1, l2_w2, l2_w4, l2_buf, stats);
  stats_kernel<<<1, 1, 0, stream>>>(stats, gamma, beta, (float)(BATCH * 49), 2, 6);
  gemm_kernel<<<BATCH / 128, 256, 0, stream>>>(l2_buf, stats, (const float2*)wswz,
                                               b_fc1, b_fc2, out);
}